// Brouwer_predict_27487790695016
// MI455X (gfx1250) — compile-verified
//
#include <hip/hip_runtime.h>
#include <hip/hip_bf16.h>

// ---------------------------------------------------------------------------
// GIN network on MI455X (gfx1250, wave32).
//   h1 = relu((segsum(x) + x) @ W1 + b1)        N x 128
//   h2 = relu((segsum(h1) + h1) @ W2 + b2)      N x 256
//   h3 = relu(h2 @ W3 + b3)                     N x 128
//   h4 = relu(h3 @ W4 + b4)                     N x 128
//   out = h4 @ W5 + b5                          N x 1
// Dense GEMMs use V_WMMA_F32_16X16X4_F32 (full fp32 precision, one wave per
// 16x16 output tile). Aggregation is L2-resident float atomics.
// ---------------------------------------------------------------------------

typedef __attribute__((ext_vector_type(2))) float v2f;
typedef __attribute__((ext_vector_type(8))) float v8f;

// ---- segment-sum scatter: agg[dst[e]] += h[src[e]] (agg pre-seeded with h) --
__global__ __launch_bounds__(256) void gin_scatter_add(
    const float* __restrict__ h, const int* __restrict__ src,
    const int* __restrict__ dst, float* __restrict__ agg,
    int F, int F4, int total) {
  int t = blockIdx.x * 256 + threadIdx.x;
  if (t >= total) return;
  int e = t / F4;
  int g = t - e * F4;
  int s = src[e];
  int d = dst[e];
  const float4 v = *(const float4*)(h + (size_t)s * F + g * 4);
  float* o = agg + (size_t)d * F + g * 4;
  atomicAdd(o + 0, v.x);
  atomicAdd(o + 1, v.y);
  atomicAdd(o + 2, v.z);
  atomicAdd(o + 3, v.w);
}

// ---- fp32 WMMA GEMM: out = act(A[MxK] @ W[KxNout] + bias) -------------------
// One wave per 16x16 output tile. M must be a multiple of 16 (50000 = 16*3125),
// K a multiple of 4, Nout a multiple of 16.
__global__ __launch_bounds__(128) void gemm_wmma_f32(
    const float* __restrict__ A, const float* __restrict__ W,
    const float* __restrict__ bias, float* __restrict__ out,
    int K, int Nout, int numNT, int totalTiles, int doRelu) {
  const int tile = blockIdx.x * 4 + (threadIdx.x >> 5);  // 4 waves / block
  if (tile >= totalTiles) return;                         // wave-uniform branch
  const int mt = tile / numNT;
  const int nt = tile - mt * numNT;
  const int lane = threadIdx.x & 31;
  const int r = lane & 15;    // row within 16 (A) / column within 16 (B,C,D)
  const int sub = lane >> 4;  // half-wave: selects K pair / output row offset

  const float* __restrict__ Arow = A + (size_t)(mt * 16 + r) * K;
  const float* __restrict__ Wcol = W + (size_t)(nt * 16 + r);

  v8f acc = {0.f, 0.f, 0.f, 0.f, 0.f, 0.f, 0.f, 0.f};

#pragma unroll 4
  for (int k = 0; k < K; k += 4) {
    const int kk = k + 2 * sub;
    // A 16x4 tile: lanes 0-15 carry K={k,k+1}, lanes 16-31 carry K={k+2,k+3}
    v2f a;
    {
      const float2 av = *(const float2*)(Arow + kk);   // 8B-aligned (kk even)
      a.x = av.x;
      a.y = av.y;
    }
    // B 4x16 tile: same half-wave split across K rows, columns across lanes
    v2f b;
    b.x = Wcol[(size_t)kk * Nout];
    b.y = Wcol[(size_t)(kk + 1) * Nout];
    acc = __builtin_amdgcn_wmma_f32_16x16x4_f32(
        /*neg_a=*/false, a, /*neg_b=*/false, b,
        /*c_mod=*/(short)0, acc, /*reuse_a=*/false, /*reuse_b=*/false);
  }

  const float bv = bias[nt * 16 + r];
  // C/D layout: VGPR v, lanes 0-15 -> M=v, lanes 16-31 -> M=v+8; N = r
  float* __restrict__ orow =
      out + (size_t)(mt * 16 + 8 * sub) * Nout + nt * 16 + r;
#pragma unroll
  for (int v = 0; v < 8; ++v) {
    float val = acc[v] + bv;
    if (doRelu) val = fmaxf(val, 0.f);
    orow[(size_t)v * Nout] = val;
  }
}

// ---- final projection: out[i] = h[i,:] . W5 + b5 (one wave per node) --------
__global__ __launch_bounds__(256) void gin_out_proj(
    const float* __restrict__ h, const float* __restrict__ W5,
    const float* __restrict__ b5, float* __restrict__ out, int M) {
  const int wave = (blockIdx.x * 256 + threadIdx.x) >> 5;
  const int lane = threadIdx.x & 31;
  if (wave >= M) return;
  const float* row = h + (size_t)wave * 128;
  float s = 0.f;
#pragma unroll
  for (int j = 0; j < 4; ++j) {
    const int k = lane + 32 * j;
    s += row[k] * W5[k];
  }
#pragma unroll
  for (int off = 16; off > 0; off >>= 1) s += __shfl_xor(s, off, 32);
  if (lane == 0) out[wave] = s + b5[0];
}

extern "C" void kernel_launch(void* const* d_in, const int* in_sizes, int n_in,
                              void* d_out, int out_size, void* d_ws,
                              size_t ws_size, hipStream_t stream) {
  const float* x   = (const float*)d_in[0];
  const int* edges = (const int*)d_in[1];
  const float* W1  = (const float*)d_in[2];
  const float* b1  = (const float*)d_in[3];
  const float* W2  = (const float*)d_in[4];
  const float* b2  = (const float*)d_in[5];
  const float* W3  = (const float*)d_in[6];
  const float* b3  = (const float*)d_in[7];
  const float* W4  = (const float*)d_in[8];
  const float* b4  = (const float*)d_in[9];
  const float* W5  = (const float*)d_in[10];
  const float* b5  = (const float*)d_in[11];
  float* out = (float*)d_out;

  const int D = 64;
  const int N = in_sizes[0] / D;   // 50000
  const int E = in_sizes[1] / 2;   // 800000
  const int* src = edges;
  const int* dst = edges + E;

  // Workspace layout (bytes), peak ~76.8 MB with reuse:
  char* ws = (char*)d_ws;
  float* agg1 = (float*)(ws + 0);                       // N*64*4  = 12.8 MB
  float* h1   = (float*)(ws + (size_t)12800000);        // N*128*4 = 25.6 MB
  float* agg2 = (float*)(ws + (size_t)51200000);        // N*128*4 = 25.6 MB
  float* h2   = (float*)(ws + 0);                       // N*256*4 = 51.2 MB (over agg1+h1, both dead)
  float* h3   = (float*)(ws + (size_t)51200000);        // N*128*4 (over agg2, dead)
  float* h4   = (float*)(ws + 0);                       // N*128*4 (over h2, dead)

  // ---- Layer 1: agg1 = x + segsum(x) ----
  hipMemcpyAsync(agg1, x, (size_t)N * 64 * sizeof(float),
                 hipMemcpyDeviceToDevice, stream);
  {
    const int F = 64, F4 = 16;
    const int total = E * F4;
    gin_scatter_add<<<(total + 255) / 256, 256, 0, stream>>>(
        x, src, dst, agg1, F, F4, total);
  }
  // h1 = relu(agg1 @ W1 + b1) : K=64, Nout=128
  {
    const int numNT = 128 / 16;
    const int tiles = (N / 16) * numNT;  // 25000
    gemm_wmma_f32<<<(tiles + 3) / 4, 128, 0, stream>>>(
        agg1, W1, b1, h1, 64, 128, numNT, tiles, 1);
  }

  // ---- Layer 2: agg2 = h1 + segsum(h1) ----
  hipMemcpyAsync(agg2, h1, (size_t)N * 128 * sizeof(float),
                 hipMemcpyDeviceToDevice, stream);
  {
    const int F = 128, F4 = 32;
    const int total = E * F4;
    gin_scatter_add<<<(total + 255) / 256, 256, 0, stream>>>(
        h1, src, dst, agg2, F, F4, total);
  }
  // h2 = relu(agg2 @ W2 + b2) : K=128, Nout=256
  {
    const int numNT = 256 / 16;
    const int tiles = (N / 16) * numNT;  // 50000
    gemm_wmma_f32<<<(tiles + 3) / 4, 128, 0, stream>>>(
        agg2, W2, b2, h2, 128, 256, numNT, tiles, 1);
  }

  // ---- h3 = relu(h2 @ W3 + b3) : K=256, Nout=128 ----
  {
    const int numNT = 128 / 16;
    const int tiles = (N / 16) * numNT;
    gemm_wmma_f32<<<(tiles + 3) / 4, 128, 0, stream>>>(
        h2, W3, b3, h3, 256, 128, numNT, tiles, 1);
  }

  // ---- h4 = relu(h3 @ W4 + b4) : K=128, Nout=128 ----
  {
    const int numNT = 128 / 16;
    const int tiles = (N / 16) * numNT;
    gemm_wmma_f32<<<(tiles + 3) / 4, 128, 0, stream>>>(
        h3, W4, b4, h4, 128, 128, numNT, tiles, 1);
  }

  // ---- out = h4 @ W5 + b5 (one wave per node) ----
  {
    const int waves_per_block = 8;
    const int blocks = (N + waves_per_block - 1) / waves_per_block;
    gin_out_proj<<<blocks, 256, 0, stream>>>(h4, W5, b5, out, N);
  }
}